// EntityLevelAggregation_88055419503365
// MI455X (gfx1250) — compile-verified
//
#include <hip/hip_runtime.h>
#include <hip/hip_bf16.h>

#define N_ENT  50000
#define N_REL  500
#define N_EDGE 800000
#define D      128
#define R      64
#define H      8
#define NEG_SLOPE 0.2f

typedef float v2f __attribute__((ext_vector_type(2)));
typedef float v8f __attribute__((ext_vector_type(8)));
typedef int   v4i __attribute__((ext_vector_type(4)));

// CDNA5 fp32 WMMA: D(16x16) = A(16x4) * B(4x16) + C
__device__ __forceinline__ v8f wmma4(v2f a, v2f b, v8f c) {
  return __builtin_amdgcn_wmma_f32_16x16x4_f32(false, a, false, b, (short)0, c,
                                               false, false);
}

__device__ __forceinline__ float leaky(float x) {
  return x >= 0.f ? x : NEG_SLOPE * x;
}

// f32 atomic max via sign-aware integer atomics (works with -inf init)
__device__ __forceinline__ void atomicMaxF32(float* addr, float val) {
  if (val >= 0.f)
    atomicMax((int*)addr, __float_as_int(val));
  else
    atomicMin((unsigned int*)addr, (unsigned int)__float_as_int(val));
}

// ---- CDNA5 async global->LDS (ASYNCcnt path); guarded fallback -------------
#if __has_builtin(__builtin_amdgcn_global_load_async_to_lds_b128)
#define HAVE_ASYNC_LDS 1
#else
#define HAVE_ASYNC_LDS 0
#endif

__device__ __forceinline__ void async_copy16(const float* g, float* l) {
#if HAVE_ASYNC_LDS
  // builtin expects v4i pointers in AS1 (global/__device__) and AS3 (LDS)
  void* gv = const_cast<float*>(g);
  __attribute__((address_space(1))) void* g1 =
      (__attribute__((address_space(1))) void*)gv;
  __attribute__((address_space(1))) v4i* g4 =
      (__attribute__((address_space(1))) v4i*)g1;
  void* lv = (void*)l;
  __attribute__((address_space(3))) void* l3 =
      (__attribute__((address_space(3))) void*)lv;
  __attribute__((address_space(3))) v4i* l4 =
      (__attribute__((address_space(3))) v4i*)l3;
  __builtin_amdgcn_global_load_async_to_lds_b128(g4, l4, 0, 0);
#else
  *(float4*)l = *(const float4*)g;
#endif
}

__device__ __forceinline__ void async_wait0() {
#if HAVE_ASYNC_LDS
#if __has_builtin(__builtin_amdgcn_s_wait_asynccnt)
  __builtin_amdgcn_s_wait_asynccnt(0);
#else
  asm volatile("s_wait_asynccnt 0" ::: "memory");
#endif
#endif
}

// ---------------------------------------------------------------- utilities
__global__ void k_zero(float* __restrict__ p, long n) {
  long i = (long)blockIdx.x * blockDim.x + threadIdx.x;
  const long stride = (long)gridDim.x * blockDim.x;
  for (; i < n; i += stride) p[i] = 0.f;
}

// ------------------------------------------------ pass 1: degree + z_bar sum
// one wave (32 lanes) per edge; lane covers 2 of the 64 relation components
__global__ void k_deg_zbar(const int* __restrict__ dstI,
                           const int* __restrict__ etI,
                           const float* __restrict__ z,
                           float* __restrict__ deg,
                           float* __restrict__ zbar) {
  const int e = blockIdx.x * 8 + (threadIdx.x >> 5);
  if (e >= N_EDGE) return;
  const int lane = threadIdx.x & 31;
  const int d = dstI[e], r = etI[e];
  if (lane == 0) atomicAdd(&deg[d], 1.0f);
  const float2 zv = *(const float2*)(z + (size_t)r * R + lane * 2);
  atomicAdd(zbar + (size_t)d * R + lane * 2 + 0, zv.x);
  atomicAdd(zbar + (size_t)d * R + lane * 2 + 1, zv.y);
}

__global__ void k_zbar_norm(float* __restrict__ zbar,
                            const float* __restrict__ deg) {
  const int t = blockIdx.x * 256 + threadIdx.x;  // N_ENT*16 threads
  if (t >= N_ENT * 16) return;
  const int i = t >> 4, c = (t & 15) * 4;
  const float s = 1.0f / fmaxf(deg[i], 1.0f);
  float4* p = (float4*)(zbar + (size_t)i * R + c);
  float4 v = *p;
  v.x *= s; v.y *= s; v.z *= s; v.w *= s;
  *p = v;
}

// ------------------------------- GEMM: [hP1|hP2|hWh] = h @ [P1|P2|Wh] (WMMA)
// block = 8 waves, 16 rows of h; each wave owns 3 of 24 (matrix, n-tile) pairs
__global__ void k_gemm_h(const float* __restrict__ h,
                         const float* __restrict__ P_w,
                         const float* __restrict__ Wc_w,
                         float* __restrict__ hP1,
                         float* __restrict__ hP2,
                         float* __restrict__ hWh) {
  __shared__ float As[16][D];
  const int t = threadIdx.x;
  const int row0 = blockIdx.x * 16;
  {
    const int r = t >> 4, c = (t & 15) * 8;
    const float* src = h + (size_t)(row0 + r) * D + c;
    async_copy16(src, &As[r][c]);          // 2 x b128 per thread = 8 KB tile
    async_copy16(src + 4, &As[r][c + 4]);
    async_wait0();
  }
  __syncthreads();
  const int lane = t & 31;
  const int w = t >> 5;
  const int mrow = lane & 15;
  const int kb = (lane >> 4) * 2;
  const int ncol = lane & 15;

  const float* Bmat[3];
  int noff[3];
  v8f acc[3];
  #pragma unroll
  for (int j = 0; j < 3; ++j) {
    const int p = w * 3 + j;            // 0..23
    const int mat = p >> 3;             // 0:P1 1:P2 2:Wh
    noff[j] = (p & 7) * 16 + ncol;
    Bmat[j] = (mat == 0) ? P_w : (mat == 1 ? (P_w + D * D) : Wc_w);
    #pragma unroll
    for (int i = 0; i < 8; ++i) acc[j][i] = 0.f;
  }
  #pragma unroll 4
  for (int k = 0; k < D; k += 4) {
    v2f a;
    a.x = As[mrow][k + kb];
    a.y = As[mrow][k + kb + 1];
    #pragma unroll
    for (int j = 0; j < 3; ++j) {
      v2f b;
      b.x = Bmat[j][(size_t)(k + kb) * D + noff[j]];
      b.y = Bmat[j][(size_t)(k + kb + 1) * D + noff[j]];
      acc[j] = wmma4(a, b, acc[j]);
    }
  }
  float* outs[3] = {hP1, hP2, hWh};
  #pragma unroll
  for (int j = 0; j < 3; ++j) {
    float* dst = outs[(w * 3 + j) >> 3];
    #pragma unroll
    for (int v = 0; v < 8; ++v) {
      const int m = v + 8 * (lane >> 4);
      dst[(size_t)(row0 + m) * D + noff[j]] = acc[j][v];
    }
  }
}

// --------- GEMM: zP3 = z@P3, zWz = z@Wz  (500 rows, padded to 512)  (WMMA)
__global__ void k_gemm_z(const float* __restrict__ z,
                         const float* __restrict__ P_w,
                         const float* __restrict__ Wc_w,
                         float* __restrict__ zP3,
                         float* __restrict__ zWz) {
  __shared__ float As[16][R];
  const int t = threadIdx.x;
  const int row0 = blockIdx.x * 16;
  {
    const int r = t >> 4, c = (t & 15) * 4;
    int rr = row0 + r;
    if (rr >= N_REL) rr = N_REL - 1;   // clamp loads; stores are guarded
    async_copy16(z + (size_t)rr * R + c, &As[r][c]);
    async_wait0();
  }
  __syncthreads();
  const int lane = t & 31;
  const int w = t >> 5;
  const int mrow = lane & 15;
  const int kb = (lane >> 4) * 2;
  const int n = w * 16 + (lane & 15);
  const float* B3 = P_w + 2 * D * D;   // P3 = P_w rows 256..319
  const float* Bz = Wc_w + D * D;      // Wz = Wc_w rows 128..191
  v8f acc3, accz;
  #pragma unroll
  for (int i = 0; i < 8; ++i) { acc3[i] = 0.f; accz[i] = 0.f; }
  #pragma unroll 4
  for (int k = 0; k < R; k += 4) {
    v2f a; a.x = As[mrow][k + kb]; a.y = As[mrow][k + kb + 1];
    v2f b;
    b.x = B3[(size_t)(k + kb) * D + n];
    b.y = B3[(size_t)(k + kb + 1) * D + n];
    acc3 = wmma4(a, b, acc3);
    b.x = Bz[(size_t)(k + kb) * D + n];
    b.y = Bz[(size_t)(k + kb + 1) * D + n];
    accz = wmma4(a, b, accz);
  }
  #pragma unroll
  for (int v = 0; v < 8; ++v) {
    const int row = row0 + v + 8 * (lane >> 4);
    if (row < N_REL) {
      zP3[(size_t)row * D + n] = acc3[v];
      zWz[(size_t)row * D + n] = accz[v];
    }
  }
}

// - GEMM z_bar@[P3|Wz] fused epilogue: pre_self -> logit_self, mx; t_self (WMMA)
__global__ void k_gemm_zbar(const float* __restrict__ zbar,
                            const float* __restrict__ P_w,
                            const float* __restrict__ Wc_w,
                            const float* __restrict__ hP1,
                            const float* __restrict__ hP2,
                            const float* __restrict__ hWh,
                            const float* __restrict__ y_w,
                            float* __restrict__ tSelf,
                            float* __restrict__ logitS,
                            float* __restrict__ mx) {
  __shared__ float As[16][R];
  __shared__ float preS[16][D];
  const int t = threadIdx.x;
  const int row0 = blockIdx.x * 16;
  {
    const int r = t >> 4, c = (t & 15) * 4;
    async_copy16(zbar + (size_t)(row0 + r) * R + c, &As[r][c]);
    async_wait0();
  }
  __syncthreads();
  const int lane = t & 31;
  const int w = t >> 5;
  const int mrow = lane & 15;
  const int kb = (lane >> 4) * 2;
  const int n = w * 16 + (lane & 15);
  const float* B3 = P_w + 2 * D * D;
  const float* Bz = Wc_w + D * D;
  v8f acc3, accz;
  #pragma unroll
  for (int i = 0; i < 8; ++i) { acc3[i] = 0.f; accz[i] = 0.f; }
  #pragma unroll 4
  for (int k = 0; k < R; k += 4) {
    v2f a; a.x = As[mrow][k + kb]; a.y = As[mrow][k + kb + 1];
    v2f b;
    b.x = B3[(size_t)(k + kb) * D + n];
    b.y = B3[(size_t)(k + kb + 1) * D + n];
    acc3 = wmma4(a, b, acc3);
    b.x = Bz[(size_t)(k + kb) * D + n];
    b.y = Bz[(size_t)(k + kb + 1) * D + n];
    accz = wmma4(a, b, accz);
  }
  #pragma unroll
  for (int v = 0; v < 8; ++v) {
    const int m = v + 8 * (lane >> 4);
    const size_t ro = (size_t)(row0 + m) * D + n;
    preS[m][n] = leaky(hP1[ro] + hP2[ro] + acc3[v]);   // leaky(pre_self)
    tSelf[ro] = hWh[ro] + accz[v];                     // t_self
  }
  __syncthreads();
  if (t < 128) {  // 16 rows x 8 heads: logit_self = leaky(pre_self) @ y_w
    const int r = t >> 3, hh = t & 7;
    float s = 0.f;
    #pragma unroll 8
    for (int c = 0; c < D; ++c) s += preS[r][c] * y_w[c * H + hh];
    logitS[(size_t)(row0 + r) * H + hh] = s;
    mx[(size_t)(row0 + r) * H + hh] = s;  // seed segment max with self logit
  }
}

// ------------------- edge logits: wave per edge, lane owns 4 of 128 channels
__global__ void k_edge_logit(const int* __restrict__ srcI,
                             const int* __restrict__ dstI,
                             const int* __restrict__ etI,
                             const float* __restrict__ hP1,
                             const float* __restrict__ hP2,
                             const float* __restrict__ zP3,
                             const float* __restrict__ hWh,
                             const float* __restrict__ y_w,
                             float* __restrict__ logitE,
                             float* __restrict__ mx) {
  __shared__ float yws[D * H];  // 4 KB
  const int t = threadIdx.x;
  ((float4*)yws)[t] = ((const float4*)y_w)[t];  // 256 * 4 = 1024 floats
  __syncthreads();
  const int e = blockIdx.x * 8 + (t >> 5);
  if (e >= N_EDGE) return;
  const int lane = t & 31;
  const int s = srcI[e], d = dstI[e], r = etI[e];
  const int c0 = lane * 4;
  // warm L2 with the t_edge gather row needed by the aggregation pass
  __builtin_prefetch(hWh + (size_t)s * D + c0, 0, 3);
  const float4 p1 = *(const float4*)(hP1 + (size_t)d * D + c0);
  const float4 p2 = *(const float4*)(hP2 + (size_t)s * D + c0);
  const float4 p3 = *(const float4*)(zP3 + (size_t)r * D + c0);
  float lv[4];
  lv[0] = leaky(p1.x + p2.x + p3.x);
  lv[1] = leaky(p1.y + p2.y + p3.y);
  lv[2] = leaky(p1.z + p2.z + p3.z);
  lv[3] = leaky(p1.w + p2.w + p3.w);
  float part[H];
  #pragma unroll
  for (int hh = 0; hh < H; ++hh) {
    float v = lv[0] * yws[(c0 + 0) * H + hh] + lv[1] * yws[(c0 + 1) * H + hh] +
              lv[2] * yws[(c0 + 2) * H + hh] + lv[3] * yws[(c0 + 3) * H + hh];
    #pragma unroll
    for (int off = 16; off > 0; off >>= 1) v += __shfl_xor(v, off, 32);
    part[hh] = v;
  }
  if (lane == 0) {
    #pragma unroll
    for (int hh = 0; hh < H; ++hh) {
      logitE[(size_t)e * H + hh] = part[hh];
      atomicMaxF32(&mx[(size_t)d * H + hh], part[hh]);
    }
  }
}

// ------------------------------------ softmax numerators / denominators
__global__ void k_node_exself(const float* __restrict__ logitS,
                              const float* __restrict__ mx,
                              float* __restrict__ exS,
                              float* __restrict__ denom) {
  const int t = blockIdx.x * 256 + threadIdx.x;
  if (t >= N_ENT * H) return;
  const float ex = expf(logitS[t] - mx[t]);
  exS[t] = ex;
  denom[t] = ex;  // plain init-write; edges atomically accumulate after
}

__global__ void k_edge_ex(const int* __restrict__ dstI,
                          float* __restrict__ logitE,  // in-place -> ex_edge
                          const float* __restrict__ mx,
                          float* __restrict__ denom) {
  const int t = blockIdx.x * 256 + threadIdx.x;
  if (t >= N_EDGE * H) return;
  const int e = t >> 3, hh = t & 7;
  const int d = dstI[e];
  const float ex = expf(logitE[t] - mx[(size_t)d * H + hh]);
  logitE[t] = ex;
  atomicAdd(&denom[(size_t)d * H + hh], ex);
}

// --------------------- message aggregation: wave per edge, 128 f32 atomics
__global__ void k_edge_agg(const int* __restrict__ srcI,
                           const int* __restrict__ dstI,
                           const int* __restrict__ etI,
                           const float* __restrict__ hWh,
                           const float* __restrict__ zWz,
                           const float* __restrict__ exE,
                           const float* __restrict__ denom,
                           float* __restrict__ agg) {
  const int e = blockIdx.x * 8 + (threadIdx.x >> 5);
  if (e >= N_EDGE) return;
  const int lane = threadIdx.x & 31;
  const int s = srcI[e], d = dstI[e], r = etI[e];
  const int c0 = lane * 4;
  const int hh = lane >> 2;  // 4 consecutive channels share one head (HD=16)
  const float alpha = exE[(size_t)e * H + hh] / denom[(size_t)d * H + hh];
  const float4 tw = *(const float4*)(hWh + (size_t)s * D + c0);
  const float4 tz = *(const float4*)(zWz + (size_t)r * D + c0);
  float* ap = agg + (size_t)d * D + c0;
  atomicAdd(ap + 0, alpha * (tw.x + tz.x));
  atomicAdd(ap + 1, alpha * (tw.y + tz.y));
  atomicAdd(ap + 2, alpha * (tw.z + tz.z));
  atomicAdd(ap + 3, alpha * (tw.w + tz.w));
}

// ------------------------------------- self message + residual + leaky + mask
__global__ void k_node_final(const float* __restrict__ h,
                             const float* __restrict__ deg,
                             const float* __restrict__ exS,
                             const float* __restrict__ denom,
                             const float* __restrict__ tSelf,
                             const float* __restrict__ agg,
                             const float* __restrict__ rw,
                             float* __restrict__ out) {
  const int idx = blockIdx.x * 256 + threadIdx.x;
  if (idx >= N_ENT * 32) return;
  const int i = idx >> 5;
  const int c0 = (idx & 31) * 4;
  const int hh = c0 >> 4;
  const float4 hv = *(const float4*)(h + (size_t)i * D + c0);
  float4 o;
  if (deg[i] > 0.f) {
    const float a = exS[(size_t)i * H + hh] / denom[(size_t)i * H + hh];
    const float rr = rw[0];
    const float4 ag = *(const float4*)(agg + (size_t)i * D + c0);
    const float4 ts = *(const float4*)(tSelf + (size_t)i * D + c0);
    o.x = leaky(ag.x + a * ts.x + rr * hv.x);
    o.y = leaky(ag.y + a * ts.y + rr * hv.y);
    o.z = leaky(ag.z + a * ts.z + rr * hv.z);
    o.w = leaky(ag.w + a * ts.w + rr * hv.w);
  } else {
    o = hv;
  }
  *(float4*)(out + (size_t)i * D + c0) = o;
}

extern "C" void kernel_launch(void* const* d_in, const int* in_sizes, int n_in,
                              void* d_out, int out_size, void* d_ws,
                              size_t ws_size, hipStream_t stream) {
  (void)in_sizes; (void)n_in; (void)out_size; (void)ws_size;
  const float* h    = (const float*)d_in[0];
  const float* z    = (const float*)d_in[1];
  const int*   ei   = (const int*)d_in[2];
  const int*   et   = (const int*)d_in[3];
  const float* Wc_w = (const float*)d_in[4];
  const float* P_w  = (const float*)d_in[5];
  const float* y_w  = (const float*)d_in[6];
  const float* rw   = (const float*)d_in[7];
  const int* srcI = ei;
  const int* dstI = ei + N_EDGE;
  float* out = (float*)d_out;

  // workspace layout (floats); agg aliases hP1 (dead after edge-logit pass),
  // ex_edge aliases logit_edge (in-place exp). ~146 MB total -> L2-resident.
  float* ws = (float*)d_ws;
  size_t o = 0;
  float* deg    = ws + o; o += 50048;
  float* zbar   = ws + o; o += (size_t)N_ENT * R;
  float* hP1    = ws + o; o += (size_t)N_ENT * D;
  float* hP2    = ws + o; o += (size_t)N_ENT * D;
  float* hWh    = ws + o; o += (size_t)N_ENT * D;
  float* tSelf  = ws + o; o += (size_t)N_ENT * D;
  float* zP3    = ws + o; o += (size_t)N_REL * D;
  float* zWz    = ws + o; o += (size_t)N_REL * D;
  float* logitS = ws + o; o += (size_t)N_ENT * H;
  float* logitE = ws + o; o += (size_t)N_EDGE * H;
  float* mx     = ws + o; o += (size_t)N_ENT * H;
  float* denom  = ws + o; o += (size_t)N_ENT * H;
  float* exS    = ws + o; o += (size_t)N_ENT * H;
  float* agg = hP1;
  float* exE = logitE;

  k_zero<<<2048, 256, 0, stream>>>(deg, (long)(50048 + (size_t)N_ENT * R));
  k_deg_zbar<<<N_EDGE / 8, 256, 0, stream>>>(dstI, et, z, deg, zbar);
  k_zbar_norm<<<(N_ENT * 16) / 256, 256, 0, stream>>>(zbar, deg);
  k_gemm_h<<<N_ENT / 16, 256, 0, stream>>>(h, P_w, Wc_w, hP1, hP2, hWh);
  k_gemm_z<<<(N_REL + 15) / 16, 256, 0, stream>>>(z, P_w, Wc_w, zP3, zWz);
  k_gemm_zbar<<<N_ENT / 16, 256, 0, stream>>>(zbar, P_w, Wc_w, hP1, hP2, hWh,
                                              y_w, tSelf, logitS, mx);
  k_edge_logit<<<N_EDGE / 8, 256, 0, stream>>>(srcI, dstI, et, hP1, hP2, zP3,
                                               hWh, y_w, logitE, mx);
  k_node_exself<<<(N_ENT * H + 255) / 256, 256, 0, stream>>>(logitS, mx, exS,
                                                             denom);
  k_edge_ex<<<(N_EDGE * H) / 256, 256, 0, stream>>>(dstI, exE, mx, denom);
  k_zero<<<4096, 256, 0, stream>>>(agg, (long)N_ENT * D);  // hP1 dead now
  k_edge_agg<<<N_EDGE / 8, 256, 0, stream>>>(srcI, dstI, et, hWh, zWz, exE,
                                             denom, agg);
  k_node_final<<<(N_ENT * 32) / 256, 256, 0, stream>>>(h, deg, exS, denom,
                                                       tSelf, agg, rw, out);
}